// StripePolynomial2d_75393855914182
// MI455X (gfx1250) — compile-verified
//
#include <hip/hip_runtime.h>
#include <stdint.h>

// ---------------------------------------------------------------------------
// StripePolynomial2d — MI455X (gfx1250, wave32)
//   out[b,c,w,h] = (1/8) * sum_i  P_i( (0.5*x + pos_i(w,h)) / 256 )
// where P_i is a 512-segment piecewise quadratic Lagrange polynomial with
// weights[i,c,:] (1025 entries, C0-continuous), and pos_i is an analytic
// rotated-stripe coordinate (POSITIONS buffer reconstructed in-kernel).
// Memory-bound: ~100MB HBM traffic; gathers served from LDS; weights staged
// with CDNA5 async global->LDS loads (GLOBAL_LOAD_ASYNC_TO_LDS_B32 +
// s_wait_asynccnt).
// ---------------------------------------------------------------------------

#ifndef __has_builtin
#define __has_builtin(x) 0
#endif

#if defined(__HIP_DEVICE_COMPILE__)
#if __has_builtin(__builtin_amdgcn_global_load_async_to_lds_b32)
#define USE_ASYNC_LDS 1
#else
#define USE_ASYNC_LDS 0
#endif
#else
#define USE_ASYNC_LDS 0
#endif

#define NPOS     8
#define NW       1025      // weights per (position, channel)
#define LDS_STR  1026      // padded stride: keeps i*LDS_STR + 2*sid 8B-aligned
#define THREADS  256
#define PPT      4         // pixels per thread
#define WH       (512 * 512)

typedef __attribute__((address_space(1))) int gint_t;
typedef __attribute__((address_space(3))) int lint_t;

// ---- compile-time position constants (float32 chain mirrors numpy) --------
constexpr double DCX[4] = {1.0, 0.9238795325112867, 0.7071067811865476,
                           0.3826834323650898};
constexpr double DCY[4] = {0.0, 0.3826834323650898, 0.7071067811865476,
                           0.9238795325112867};

constexpr float fcx(int i)  { return (float)DCX[i >> 1]; }
constexpr float fcys(int i) { return (i & 1) ? -(float)DCY[i >> 1]
                                             :  (float)DCY[i >> 1]; }
constexpr float frmin(int i){ return (i & 1) ? -((float)DCY[i >> 1] * 511.0f)
                                             : 0.0f; }
constexpr float fdr(int i)  {
  float cx = (float)DCX[i >> 1], cy = (float)DCY[i >> 1];
  float rmax = (i & 1) ? (cx * 511.0f) : (cx * 511.0f + cy * 511.0f);
  return rmax - frmin(i);
}

constexpr float CXF[NPOS]  = {fcx(0),  fcx(1),  fcx(2),  fcx(3),
                              fcx(4),  fcx(5),  fcx(6),  fcx(7)};
constexpr float CYS[NPOS]  = {fcys(0), fcys(1), fcys(2), fcys(3),
                              fcys(4), fcys(5), fcys(6), fcys(7)};
constexpr float RMIN[NPOS] = {frmin(0),frmin(1),frmin(2),frmin(3),
                              frmin(4),frmin(5),frmin(6),frmin(7)};
constexpr float DRF[NPOS]  = {fdr(0),  fdr(1),  fdr(2),  fdr(3),
                              fdr(4),  fdr(5),  fdr(6),  fdr(7)};
constexpr float RATIOF = (float)(512.0 / 513.0);

__global__ __launch_bounds__(THREADS)
void stripe_poly_kernel(const float* __restrict__ x,
                        const float* __restrict__ wts,
                        float* __restrict__ out)
{
  __shared__ __align__(16) float lw[NPOS * LDS_STR];

  const int c   = blockIdx.y;     // channel 0..2
  const int tid = threadIdx.x;

  // ---- stage this channel's 8 weight tables into LDS ----------------------
#if USE_ASYNC_LDS
  for (int idx = tid; idx < NPOS * NW; idx += THREADS) {
    const int i = idx / NW;
    const int j = idx - i * NW;
    const float* src = wts + (size_t)(i * 3 + c) * NW + j;
    const uint32_t ldsOff = (uint32_t)(uintptr_t)(&lw[i * LDS_STR + j]);
    __builtin_amdgcn_global_load_async_to_lds_b32(
        (gint_t*)(uintptr_t)src,
        (lint_t*)ldsOff,
        /*offset=*/0, /*cpol=*/0);
  }
#if __has_builtin(__builtin_amdgcn_s_wait_asynccnt)
  __builtin_amdgcn_s_wait_asynccnt(0);
#else
  asm volatile("s_wait_asynccnt 0" ::: "memory");
#endif
#else
  for (int idx = tid; idx < NPOS * NW; idx += THREADS) {
    const int i = idx / NW;
    const int j = idx - i * NW;
    lw[i * LDS_STR + j] = wts[(size_t)(i * 3 + c) * NW + j];
  }
#endif
  __syncthreads();

  // ---- main streaming loop ------------------------------------------------
  const int pixBase = blockIdx.x * (THREADS * PPT);

  for (int pp = 0; pp < PPT; ++pp) {
    const int p  = pixBase + pp * THREADS + tid;      // 0 .. 512*512-1
    const float fw = (float)(p >> 9);                 // pixel x-index
    const float fh = (float)(p & 511);                // pixel y-index

    // analytic rotated-stripe coordinate per position (shared across batches)
    float pos[NPOS];
#pragma unroll
    for (int i = 0; i < NPOS; ++i) {
      float r = CXF[i] * fw + CYS[i] * fh;
      float q = ((r - RMIN[i]) * RATIOF) / DRF[i];
      pos[i] = (q - 0.5f) * 512.0f;
    }

    const float* xp = x   + ((size_t)c << 18) + p;    // b-stride = 3<<18
    float*       op = out + ((size_t)c << 18) + p;

#pragma unroll 4
    for (int b = 0; b < 16; ++b) {
      const float xv = xp[(size_t)b * (3u << 18)];
      float acc = 0.0f;
#pragma unroll
      for (int i = 0; i < NPOS; ++i) {
        // xs = ((0.5x+pos)/256 + 1)*256  ==  0.5x + pos + 256  (exact pow2)
        float xs = fmaf(0.5f, xv, pos[i]) + 256.0f;
        float sf = floorf(xs);
        sf = fminf(fmaxf(sf, 0.0f), 511.0f);
        const int sid = (int)sf;
        const float t = 2.0f * (xs - sf) - 1.0f;

        // gather 3 consecutive weights: base is 8B-aligned (ds_load_b64+b32)
        const float* wp = &lw[i * LDS_STR + 2 * sid];
        const float2 w01 = *reinterpret_cast<const float2*>(wp);
        const float  w2  = wp[2];

        const float b0 = 0.5f * t * (t - 1.0f);   // Lagrange @ {-1,0,1}
        const float b1 = 1.0f - t * t;
        const float b2 = 0.5f * t * (t + 1.0f);

        acc += b0 * w01.x + b1 * w01.y + b2 * w2;
      }
      op[(size_t)b * (3u << 18)] = acc * 0.125f;
    }
  }
}

extern "C" void kernel_launch(void* const* d_in, const int* in_sizes, int n_in,
                              void* d_out, int out_size, void* d_ws, size_t ws_size,
                              hipStream_t stream) {
  (void)in_sizes; (void)n_in; (void)out_size; (void)d_ws; (void)ws_size;
  const float* x   = (const float*)d_in[0];   // [16,3,512,512] f32
  const float* wts = (const float*)d_in[1];   // [8,3,1025] f32
  float*       out = (float*)d_out;           // [16,3,512,512] f32

  dim3 block(THREADS, 1, 1);
  dim3 grid(WH / (THREADS * PPT), 3, 1);      // 256 x 3 blocks
  stripe_poly_kernel<<<grid, block, 0, stream>>>(x, wts, out);
}